// GraphSAGE_16965120819650
// MI455X (gfx1250) — compile-verified
//
#include <hip/hip_runtime.h>

typedef __attribute__((ext_vector_type(2))) float v2f;
typedef __attribute__((ext_vector_type(8))) float v8f;

#define CH 64

// ---------------- utility: zero fill ----------------
__global__ void zero_f32(float* __restrict__ p, int n) {
    int i = blockIdx.x * blockDim.x + threadIdx.x;
    if (i < n) p[i] = 0.0f;
}

// ---------------- degree: one thread per edge ----------------
__global__ void degree_kernel(const long long* __restrict__ dst,
                              float* __restrict__ deg, int n_edges) {
    int e = blockIdx.x * blockDim.x + threadIdx.x;
    if (e < n_edges) atomicAdd(&deg[(int)dst[e]], 1.0f);
}

// ------------- scatter-add: thread per (edge, channel) -------------
// agg[dst[e]][c] += h[src[e]][c]; lanes cover consecutive channels -> coalesced
__global__ void scatter_kernel(const float* __restrict__ h,
                               const long long* __restrict__ src,
                               const long long* __restrict__ dst,
                               float* __restrict__ agg, int n_edges) {
    long long gid = (long long)blockIdx.x * blockDim.x + threadIdx.x;
    int e = (int)(gid >> 6);
    int c = (int)(gid & 63);
    if (e < n_edges) {
        int s = (int)src[e];
        int d = (int)dst[e];
        atomicAdd(&agg[(long long)d * CH + c], h[(long long)s * CH + c]);
    }
}

// ------------- SAGE layer 1: h = relu(mean@Wl.T + b + x@Wr.T) -------------
// One wave per 16-node tile. fp32 WMMA 16x16x4.
// A layout: lane l, vgpr j -> A[l%16][kbase + 2*(l/16) + j]
// B layout: lane l, vgpr j -> B[kbase + j + 2*(l/16)][l%16] ; B = W^T tile,
//           so B element = W[obase + l%16][kbase + 2*(l/16) + j]  (float2 load)
// D layout: lane l, vgpr r -> out[node_base + r + 8*(l/16)][obase + l%16]
__global__ void sage1_wmma(const float* __restrict__ x,
                           const float* __restrict__ agg,
                           const float* __restrict__ deg,
                           const float* __restrict__ Wl,   // [64][64]
                           const float* __restrict__ Wr,   // [64][64]
                           const float* __restrict__ bias, // [64]
                           float* __restrict__ h, int n_nodes) {
    const int lane  = threadIdx.x & 31;
    const int wave  = blockIdx.x * (blockDim.x >> 5) + (threadIdx.x >> 5);
    const int node_base = wave * 16;
    if (node_base >= n_nodes) return;               // wave-uniform

    const int m     = lane & 15;                    // row within tile / out col
    const int khalf = lane >> 4;                    // 0 or 1

    int node = node_base + m;
    if (node > n_nodes - 1) node = n_nodes - 1;     // clamp (safe duplicate)
    const float inv = 1.0f / fmaxf(deg[node], 1.0f);
    const float* __restrict__ xrow = x   + (long long)node * CH;
    const float* __restrict__ arow = agg + (long long)node * CH;

    for (int nt = 0; nt < 4; ++nt) {                // 4 output column tiles
        const int obase = nt * 16;
        const float bv = bias[obase + m];
        v8f acc = { bv, bv, bv, bv, bv, bv, bv, bv };
        const float* __restrict__ wl_row = Wl + (long long)(obase + m) * CH;
        const float* __restrict__ wr_row = Wr + (long long)(obase + m) * CH;
#pragma unroll
        for (int k = 0; k < CH; k += 4) {
            const int ka = k + 2 * khalf;
            v2f a_mean;
            a_mean.x = arow[ka]     * inv;
            a_mean.y = arow[ka + 1] * inv;
            v2f a_x = *(const v2f*)(xrow + ka);
            v2f b_l = *(const v2f*)(wl_row + ka);
            v2f b_r = *(const v2f*)(wr_row + ka);
            acc = __builtin_amdgcn_wmma_f32_16x16x4_f32(
                false, a_mean, false, b_l, (short)0, acc, false, false);
            acc = __builtin_amdgcn_wmma_f32_16x16x4_f32(
                false, a_x,    false, b_r, (short)0, acc, false, false);
        }
#pragma unroll
        for (int r = 0; r < 8; ++r) {
            const int out_node = node_base + r + 8 * khalf;
            if (out_node < n_nodes)
                h[(long long)out_node * CH + obase + m] = fmaxf(acc[r], 0.0f);
        }
    }
}

// ------------- SAGE layer 2: out = mean2@W2l.T + b2 + h@W2r.T (OUT=2) -------------
// Same tiling; only output columns 0..1 are real. Lanes with m>=2 feed zero
// B rows (branch reconverges before each WMMA, EXEC all-1s at WMMA).
__global__ void sage2_wmma(const float* __restrict__ h,
                           const float* __restrict__ agg,
                           const float* __restrict__ deg,
                           const float* __restrict__ Wl,   // [2][64]
                           const float* __restrict__ Wr,   // [2][64]
                           const float* __restrict__ bias, // [2]
                           float* __restrict__ out, int n_nodes) {
    const int lane  = threadIdx.x & 31;
    const int wave  = blockIdx.x * (blockDim.x >> 5) + (threadIdx.x >> 5);
    const int node_base = wave * 16;
    if (node_base >= n_nodes) return;               // wave-uniform

    const int m     = lane & 15;
    const int khalf = lane >> 4;

    int node = node_base + m;
    if (node > n_nodes - 1) node = n_nodes - 1;
    const float inv = 1.0f / fmaxf(deg[node], 1.0f);
    const float* __restrict__ hrow = h   + (long long)node * CH;
    const float* __restrict__ arow = agg + (long long)node * CH;

    const bool valid = (m < 2);
    const float bv = valid ? bias[m] : 0.0f;
    v8f acc = { bv, bv, bv, bv, bv, bv, bv, bv };

#pragma unroll
    for (int k = 0; k < CH; k += 4) {
        const int ka = k + 2 * khalf;
        v2f a_mean;
        a_mean.x = arow[ka]     * inv;
        a_mean.y = arow[ka + 1] * inv;
        v2f a_h = *(const v2f*)(hrow + ka);
        v2f b_l = { 0.0f, 0.0f };
        v2f b_r = { 0.0f, 0.0f };
        if (valid) {
            b_l = *(const v2f*)(Wl + m * CH + ka);
            b_r = *(const v2f*)(Wr + m * CH + ka);
        }
        acc = __builtin_amdgcn_wmma_f32_16x16x4_f32(
            false, a_mean, false, b_l, (short)0, acc, false, false);
        acc = __builtin_amdgcn_wmma_f32_16x16x4_f32(
            false, a_h,    false, b_r, (short)0, acc, false, false);
    }

    if (valid) {
#pragma unroll
        for (int r = 0; r < 8; ++r) {
            const int out_node = node_base + r + 8 * khalf;
            if (out_node < n_nodes)
                out[(long long)out_node * 2 + m] = acc[r];
        }
    }
}

extern "C" void kernel_launch(void* const* d_in, const int* in_sizes, int n_in,
                              void* d_out, int out_size, void* d_ws, size_t ws_size,
                              hipStream_t stream) {
    const float*     x   = (const float*)d_in[0];
    const long long* ei  = (const long long*)d_in[1];   // int64 [2, E]
    const float*     W1l = (const float*)d_in[2];
    const float*     W1r = (const float*)d_in[3];
    const float*     b1  = (const float*)d_in[4];
    const float*     W2l = (const float*)d_in[5];
    const float*     W2r = (const float*)d_in[6];
    const float*     b2  = (const float*)d_in[7];

    const int n_nodes = in_sizes[0] / CH;
    const int n_edges = in_sizes[1] / 2;
    const long long* src = ei;
    const long long* dst = ei + n_edges;

    // workspace layout (floats): deg[N] | agg[N*64] | h[N*64]
    float* deg = (float*)d_ws;
    float* agg = deg + n_nodes;
    float* hbuf = agg + (size_t)n_nodes * CH;

    // zero deg + agg (contiguous region)
    const int nz = n_nodes * (CH + 1);
    zero_f32<<<(nz + 255) / 256, 256, 0, stream>>>(deg, nz);

    // degree (computed once; identical for both layers)
    degree_kernel<<<(n_edges + 255) / 256, 256, 0, stream>>>(dst, deg, n_edges);

    // layer-1 scatter: agg += x[src]
    const long long sc_threads = (long long)n_edges * CH;
    const int sc_blocks = (int)((sc_threads + 255) / 256);
    scatter_kernel<<<sc_blocks, 256, 0, stream>>>(x, src, dst, agg, n_edges);

    // layer-1 fused GEMM+bias+ReLU (WMMA f32)
    const int tiles = (n_nodes + 15) / 16;
    const int waves_per_block = 8;                    // 256 threads = 8 waves
    const int s_blocks = (tiles + waves_per_block - 1) / waves_per_block;
    sage1_wmma<<<s_blocks, 256, 0, stream>>>(x, agg, deg, W1l, W1r, b1, hbuf, n_nodes);

    // re-zero agg for layer 2
    const int na = n_nodes * CH;
    zero_f32<<<(na + 255) / 256, 256, 0, stream>>>(agg, na);

    // layer-2 scatter: agg += h[src]
    scatter_kernel<<<sc_blocks, 256, 0, stream>>>(hbuf, src, dst, agg, n_edges);

    // layer-2 fused GEMM+bias (WMMA f32) -> d_out [N, 2]
    sage2_wmma<<<s_blocks, 256, 0, stream>>>(hbuf, agg, deg, W2l, W2r, b2,
                                             (float*)d_out, n_nodes);
}